// Dilation2D_51462298141250
// MI455X (gfx1250) — compile-verified
//
#include <hip/hip_runtime.h>

// Iterated 15x15 ones-kernel "soft dilation" (conv + bias + clip), fp32.
// Separable: 15x15 ones = (1x15) o (15x1), exact. Each 1-D ones conv is a
// banded matmul chain on V_WMMA_F32_16X16X4_F32 (fp32-exact WMMA shape).
// Memory-bound: ~2.7 GB total traffic; 134 MB working set lives in 192MB L2.
// Tile staging uses GLOBAL_LOAD_ASYNC_TO_LDS_B128 (ASYNCcnt path) when the
// toolchain exposes the builtin; band operands are hoisted into registers.

typedef __attribute__((ext_vector_type(2))) float v2f;
typedef __attribute__((ext_vector_type(8))) float v8f;

#if defined(__has_builtin)
#if __has_builtin(__builtin_amdgcn_global_load_async_to_lds_b128)
#define HAVE_ASYNC_LDS 1
#endif
#endif

#ifdef HAVE_ASYNC_LDS
// The builtin's pointer params are 16-byte int vectors (per hipcc diagnostic):
//   param0: global (AS1) v4i*, param1: LDS (AS3) v4i*, then (int offset, int cpol)
typedef int vi4 __attribute__((vector_size(4 * sizeof(int))));
typedef __attribute__((address_space(1))) vi4 g_vi4;
typedef __attribute__((address_space(3))) vi4 l_vi4;
#endif

namespace {
constexpr int IMG      = 2048;
constexpr int NIMG     = 8;
constexpr int OUT_W    = 128;          // block output tile width
constexpr int OUT_H    = 64;           // block output tile height
constexpr int IN_W     = OUT_W + 16;   // 144: cols c0-8 .. c0+135
constexpr int IN_H     = OUT_H + 32;   // 96 : rows r0-16 .. r0+79
constexpr int IN_PITCH = 148;          // 16B-aligned rows; conflict-free A reads
constexpr int H_PITCH  = 140;          // 2P=24, 8P=32 (mod 64 banks): no conflicts
constexpr int CHUNKS   = IN_W / 4;     // 36 x 16B chunks per tile row
constexpr int NCOPY    = IN_H * CHUNKS;
constexpr int ITER     = 10;
}

// 1.0 iff j+1 <= i <= j+15 : the 15-tap ones band for a window based at -8.
__device__ __forceinline__ float band01(int i, int j) {
  return ((unsigned)(i - j - 1) <= 14u) ? 1.0f : 0.0f;
}

__device__ __forceinline__ v8f wmma4(v2f a, v2f b, v8f c) {
  // (neg_a, A, neg_b, B, c_mod, C, reuse_a, reuse_b)
  return __builtin_amdgcn_wmma_f32_16x16x4_f32(false, a, false, b, (short)0, c,
                                               false, false);
}

extern "C" __global__ __launch_bounds__(256)
void dilate_iter_kernel(const float* __restrict__ in, float* __restrict__ out,
                        const float* __restrict__ bptr) {
  extern __shared__ float smem[];
  float* s_in = smem;                    // IN_H x IN_PITCH input tile (+halo)
  float* s_h  = smem + IN_H * IN_PITCH;  // IN_H x H_PITCH horizontal sums

  const int tid  = threadIdx.x;
  const int lane = tid & 31;
  const int wv   = tid >> 5;            // wave id 0..7 (wave32)
  const int lm   = lane & 15;           // M (A rows) / N (B cols) lane index
  const int koff = (lane >> 4) << 1;    // K sub-offset: lanes 16-31 hold K+2
  const int hi8  = (lane >> 4) << 3;    // C/D rows: lanes 16-31 hold M+8

  const int c0 = blockIdx.x * OUT_W;
  const int r0 = blockIdx.y * OUT_H;
  const size_t base = (size_t)blockIdx.z * IMG * IMG;

  const float bias = bptr[0];

  // ---- stage haloed input tile into LDS ------------------------------
#ifdef HAVE_ASYNC_LDS
  // Async direct-to-LDS 16B copies (bypass VGPRs, tracked by ASYNCcnt).
  // Every 16B chunk is either fully inside the image or pure zero padding
  // (column window and image bounds are both multiples of 4 floats).
  {
    for (int idx = tid; idx < NCOPY; idx += 256) {
      int rr = idx / CHUNKS;
      int cj = idx - rr * CHUNKS;
      int gr = r0 - 16 + rr;
      int gc = c0 - 8 + cj * 4;
      float* dgen = s_in + rr * IN_PITCH + cj * 4;
      if ((unsigned)gr < (unsigned)IMG && (unsigned)gc < (unsigned)IMG) {
        const float* srcp = in + base + (size_t)gr * IMG + gc;
        __builtin_amdgcn_global_load_async_to_lds_b128(
            (g_vi4*)srcp, (l_vi4*)dgen, 0, 0);
      } else {
        float4 z = make_float4(0.f, 0.f, 0.f, 0.f);
        *(float4*)dgen = z;
      }
    }
    asm volatile("s_wait_asynccnt 0" ::: "memory");
  }
#else
  for (int idx = tid; idx < IN_H * IN_W; idx += 256) {
    int rr = idx / IN_W;
    int cc = idx - rr * IN_W;
    int gr = r0 - 16 + rr;
    int gc = c0 - 8 + cc;
    float v = 0.0f;
    if ((unsigned)gr < (unsigned)IMG && (unsigned)gc < (unsigned)IMG)
      v = in[base + (size_t)gr * IMG + gc];
    s_in[rr * IN_PITCH + cc] = v;
  }
#endif
  __syncthreads();

  // ---- band fragments: computed once, reused by BOTH passes ----------
  // Horizontal pass uses band[kc] as the B operand (K x N), vertical pass
  // uses it as the A operand (M x K); the lane mapping is identical.
  v2f band[8];
#pragma unroll
  for (int kc = 0; kc < 8; ++kc) {
    const int i0 = kc * 4 + koff;
    band[kc].x = band01(i0, lm);
    band[kc].y = band01(i0 + 1, lm);
  }

  const int C = wv * 16;  // this wave's private 16-column strip

  // ---- horizontal pass: h(16x16) = In(16x32) x Band(32x16) -----------
  // Two accumulators in flight to break the WMMA C->C serial chain.
#pragma unroll
  for (int p = 0; p < 3; ++p) {
    const int R0 = (2 * p) * 16;
    const int R1 = R0 + 16;
    v8f acc0 = {}, acc1 = {};
#pragma unroll
    for (int kc = 0; kc < 8; ++kc) {
      const int i0 = kc * 4 + koff;
      v2f a0, a1;
      a0.x = s_in[(R0 + lm) * IN_PITCH + C + i0];
      a0.y = s_in[(R0 + lm) * IN_PITCH + C + i0 + 1];
      a1.x = s_in[(R1 + lm) * IN_PITCH + C + i0];
      a1.y = s_in[(R1 + lm) * IN_PITCH + C + i0 + 1];
      acc0 = wmma4(a0, band[kc], acc0);
      acc1 = wmma4(a1, band[kc], acc1);
    }
#pragma unroll
    for (int v = 0; v < 8; ++v) {
      s_h[(R0 + hi8 + v) * H_PITCH + C + lm] = acc0[v];
      s_h[(R1 + hi8 + v) * H_PITCH + C + lm] = acc1[v];
    }
  }
  // Each wave consumes only the h-columns it produced itself, so no
  // cross-wave barrier is needed; DScnt waits cover the LDS RAW.

  // ---- vertical pass: out(16x16) = Band(16x32) x H(32x16), bias+clip --
#pragma unroll
  for (int p = 0; p < 2; ++p) {
    const int Y0 = (2 * p) * 16;
    const int Y1 = Y0 + 16;
    v8f acc0 = {}, acc1 = {};
#pragma unroll
    for (int kc = 0; kc < 8; ++kc) {
      const int i0 = kc * 4 + koff;
      v2f b0, b1;
      b0.x = s_h[(Y0 + 8 + i0) * H_PITCH + C + lm];
      b0.y = s_h[(Y0 + 9 + i0) * H_PITCH + C + lm];
      b1.x = s_h[(Y1 + 8 + i0) * H_PITCH + C + lm];
      b1.y = s_h[(Y1 + 9 + i0) * H_PITCH + C + lm];
      acc0 = wmma4(band[kc], b0, acc0);
      acc1 = wmma4(band[kc], b1, acc1);
    }
#pragma unroll
    for (int v = 0; v < 8; ++v) {
      float y0 = fminf(fmaxf(acc0[v] + bias, 0.0f), 1.0f);
      float y1 = fminf(fmaxf(acc1[v] + bias, 0.0f), 1.0f);
      out[base + (size_t)(r0 + Y0 + hi8 + v) * IMG + (c0 + C + lm)] = y0;
      out[base + (size_t)(r0 + Y1 + hi8 + v) * IMG + (c0 + C + lm)] = y1;
    }
  }
}

extern "C" void kernel_launch(void* const* d_in, const int* in_sizes, int n_in,
                              void* d_out, int out_size, void* d_ws, size_t ws_size,
                              hipStream_t stream) {
  (void)in_sizes; (void)n_in; (void)out_size; (void)ws_size;
  const float* x    = (const float*)d_in[0];  // [8,1,2048,2048] fp32
  // d_in[1] = ones kernel (folded into the banded matmul), d_in[2] = bias
  const float* bptr = (const float*)d_in[2];
  float* out = (float*)d_out;
  float* ws  = (float*)d_ws;   // needs >= 134 MB: ping-pong buffer

  dim3 grid(IMG / OUT_W, IMG / OUT_H, NIMG);  // 16 x 32 x 8 blocks
  dim3 block(256);
  size_t shmem = (size_t)(IN_H * IN_PITCH + IN_H * H_PITCH) * sizeof(float);

  // 10 iterations, ping-pong ws <-> out; iteration 9 (last) writes d_out.
  const float* src = x;
  for (int it = 0; it < ITER; ++it) {
    float* dst = (it & 1) ? out : ws;
    dilate_iter_kernel<<<grid, block, shmem, stream>>>(src, dst, bptr);
    src = dst;
  }
}